// EfficientFi_51728586113088
// MI455X (gfx1250) — compile-verified
//
#include <hip/hip_runtime.h>

typedef float v2f __attribute__((ext_vector_type(2)));
typedef float v8f __attribute__((ext_vector_type(8)));
typedef unsigned int u32x4 __attribute__((ext_vector_type(4)));
typedef int i32x4 __attribute__((ext_vector_type(4)));
typedef int i32x8 __attribute__((ext_vector_type(8)));

// ---------------- enc1: conv 3->32 (3x3 pad1) + relu + maxpool2x2 ----------------
__global__ void __launch_bounds__(256) k_enc1(const float* __restrict__ x,
                                              const float* __restrict__ w,
                                              const float* __restrict__ bias,
                                              float* __restrict__ p1,
                                              int* __restrict__ idx1) {
  int t = blockIdx.x * blockDim.x + threadIdx.x;   // 1024*32*100
  if (t >= 1024 * 32 * 100) return;
  int pos = t % 100, c = (t / 100) % 32, b = t / 3200;
  int py = pos / 10, px = pos % 10;
  const float* xb = x + b * (3 * 400);
  float best = -3.4e38f; int bi = 0;
  for (int k = 0; k < 4; ++k) {
    int hy = 2 * py + (k >> 1), hx = 2 * px + (k & 1);
    float acc = bias[c];
    for (int ic = 0; ic < 3; ++ic)
      for (int ky = 0; ky < 3; ++ky) {
        int yy = hy + ky - 1;
        if (yy < 0 || yy >= 20) continue;
        for (int kx = 0; kx < 3; ++kx) {
          int xx = hx + kx - 1;
          if (xx < 0 || xx >= 20) continue;
          acc += xb[ic * 400 + yy * 20 + xx] * w[((c * 3 + ic) * 3 + ky) * 3 + kx];
        }
      }
    float r = acc > 0.f ? acc : 0.f;
    if (r > best) { best = r; bi = k; }   // strict '>' => first max (jnp.argmax)
  }
  p1[t] = best; idx1[t] = bi;
}

// ---------------- enc2: conv 32->96 (3x3 pad1) + relu + maxpool2x2 ----------------
__global__ void __launch_bounds__(256) k_enc2(const float* __restrict__ p1,
                                              const float* __restrict__ w,
                                              const float* __restrict__ bias,
                                              float* __restrict__ z,
                                              int* __restrict__ idx2) {
  __shared__ float sp[32 * 100];
  int b = blockIdx.x, tid = threadIdx.x;
  for (int i = tid; i < 3200; i += 256) sp[i] = p1[b * 3200 + i];
  __syncthreads();
  for (int o = tid; o < 96 * 25; o += 256) {
    int pos = o % 25, c = o / 25;
    int py = pos / 5, px = pos % 5;
    float best = -3.4e38f; int bi = 0;
    for (int k = 0; k < 4; ++k) {
      int hy = 2 * py + (k >> 1), hx = 2 * px + (k & 1);
      float acc = bias[c];
      for (int ic = 0; ic < 32; ++ic) {
        const float* sr = sp + ic * 100;
        const float* wr = w + (c * 32 + ic) * 9;
        for (int ky = 0; ky < 3; ++ky) {
          int yy = hy + ky - 1;
          if (yy < 0 || yy >= 10) continue;
          for (int kx = 0; kx < 3; ++kx) {
            int xx = hx + kx - 1;
            if (xx < 0 || xx >= 10) continue;
            acc += sr[yy * 10 + xx] * wr[ky * 3 + kx];
          }
        }
      }
      float r = acc > 0.f ? acc : 0.f;
      if (r > best) { best = r; bi = k; }
    }
    z[b * 2400 + o] = best; idx2[b * 2400 + o] = bi;
  }
}

// ---------------- pre-VQ 1x1 conv 96->128 ----------------
__global__ void __launch_bounds__(256) k_pre(const float* __restrict__ z,
                                             const float* __restrict__ w,
                                             const float* __restrict__ bias,
                                             float* __restrict__ vq) {
  int t = blockIdx.x * blockDim.x + threadIdx.x;   // 1024*128*25
  if (t >= 1024 * 128 * 25) return;
  int p = t % 25, e = (t / 25) % 128, b = t / 3200;
  const float* zb = z + b * 2400 + p;
  const float* wr = w + e * 96;
  float acc = bias[e];
  for (int c = 0; c < 96; ++c) acc += zb[c * 25] * wr[c];
  vq[t] = acc;
}

// ---------------- codebook squared norms ----------------
__global__ void k_cbsq(const float* __restrict__ cb, float* __restrict__ cbsq) {
  int j = blockIdx.x * blockDim.x + threadIdx.x;
  if (j >= 1024) return;
  float s = 0.f;
  for (int k = 0; k < 25; ++k) { float v = cb[j * 25 + k]; s += v * v; }
  cbsq[j] = s;
}

// ---------------- VQ argmin via V_WMMA_F32_16X16X4_F32 + TDM codebook stage ------
// wave handles 16 rows of flat; loops 64 tiles of 16 codes; K=25 padded to 28.
// Codebook (102,400 B) is DMA'd global->LDS by the Tensor Data Mover (1-D D#).
__global__ void __launch_bounds__(256) k_vq_argmin(const float* __restrict__ vq,
                                                   const float* __restrict__ cb,
                                                   const float* __restrict__ cbsq,
                                                   int* __restrict__ indices) {
  __shared__ float scb[25600];
  __shared__ float ssq[1024];
  int tid = threadIdx.x;

  // ---- TDM: tensor_load_to_lds of the full codebook (25600 x f32, 1-D tile) ----
  if (tid < 32) {
    unsigned long long ga = (unsigned long long)(const void*)cb;
    unsigned lds_base = (unsigned)(unsigned long long)(const void*)&scb[0];
    u32x4 g0;
    g0.x = 1u;                                         // count=1, user desc
    g0.y = lds_base;                                   // lds_addr (bytes)
    g0.z = (unsigned)(ga & 0xffffffffu);               // global_addr[31:0]
    g0.w = (unsigned)((ga >> 32) & 0x01ffffffu) | (2u << 30);  // addr[56:32]|type=2
    i32x8 g1;
    g1.s0 = 0x20000;        // workgroup_mask=0, data_size=2 (4 bytes)
    g1.s1 = 0x64000000;     // tensor_dim0[15:0]=25600 in [31:16]
    g1.s2 = 0x00010000;     // tensor_dim0[31:16]=0, tensor_dim1=1
    g1.s3 = 0x64000000;     // tile_dim0=25600 in [31:16]
    g1.s4 = 0;              // tile_dim1=0 (unused), tile_dim2=0
    g1.s5 = 25600;          // tensor_dim0_stride[31:0]
    g1.s6 = 0;              // stride[47:32]=0, dim1_stride lo=0
    g1.s7 = 0;              // dim1_stride hi=0
    i32x4 gz = {0, 0, 0, 0};
#if __clang_major__ >= 23
    i32x8 gz8 = {0, 0, 0, 0, 0, 0, 0, 0};
    __builtin_amdgcn_tensor_load_to_lds(g0, g1, gz, gz, gz8, 0);
#else
    __builtin_amdgcn_tensor_load_to_lds(g0, g1, gz, gz, 0);
#endif
  }
  for (int i = tid; i < 1024; i += 256) ssq[i] = cbsq[i];
  __builtin_amdgcn_s_wait_tensorcnt(0);
  __syncthreads();

  int lane = tid & 31, wave = tid >> 5;
  int row_base = blockIdx.x * 128 + wave * 16;
  int khalf = (lane < 16) ? 0 : 2;
  int row = row_base + (lane & 15);
  const float* vr = vq + row * 25;

  v2f a[7];
#pragma unroll
  for (int s = 0; s < 7; ++s) {
    int k = 4 * s + khalf;
    a[s].x = (k < 25) ? vr[k] : 0.f;
    a[s].y = (k + 1 < 25) ? vr[k + 1] : 0.f;
  }

  float best[8]; int bidx[8];
#pragma unroll
  for (int v = 0; v < 8; ++v) { best[v] = 3.4e38f; bidx[v] = 0; }

  for (int nb = 0; nb < 64; ++nb) {
    int n = nb * 16 + (lane & 15);
    const float* cr = scb + n * 25;
    v2f bb[7];
#pragma unroll
    for (int s = 0; s < 7; ++s) {
      int k = 4 * s + khalf;
      bb[s].x = (k < 25) ? cr[k] : 0.f;
      bb[s].y = (k + 1 < 25) ? cr[k + 1] : 0.f;
    }
    // two independent accumulator chains (even/odd K-steps) -> half the RAW depth
    v8f accE = {0.f, 0.f, 0.f, 0.f, 0.f, 0.f, 0.f, 0.f};
    v8f accO = {0.f, 0.f, 0.f, 0.f, 0.f, 0.f, 0.f, 0.f};
    accE = __builtin_amdgcn_wmma_f32_16x16x4_f32(false, a[0], false, bb[0], (short)0, accE, false, false);
    accO = __builtin_amdgcn_wmma_f32_16x16x4_f32(false, a[1], false, bb[1], (short)0, accO, false, false);
    accE = __builtin_amdgcn_wmma_f32_16x16x4_f32(false, a[2], false, bb[2], (short)0, accE, false, false);
    accO = __builtin_amdgcn_wmma_f32_16x16x4_f32(false, a[3], false, bb[3], (short)0, accO, false, false);
    accE = __builtin_amdgcn_wmma_f32_16x16x4_f32(false, a[4], false, bb[4], (short)0, accE, false, false);
    accO = __builtin_amdgcn_wmma_f32_16x16x4_f32(false, a[5], false, bb[5], (short)0, accO, false, false);
    accE = __builtin_amdgcn_wmma_f32_16x16x4_f32(false, a[6], false, bb[6], (short)0, accE, false, false);
    float cs = ssq[n];
#pragma unroll
    for (int v = 0; v < 8; ++v) {
      float d = cs - 2.0f * (accE[v] + accO[v]);  // ||flat||^2 const per row: argmin-safe
      if (d < best[v]) { best[v] = d; bidx[v] = n; }
    }
  }

#pragma unroll
  for (int v = 0; v < 8; ++v) {
    float bd = best[v]; int bi = bidx[v];
#pragma unroll
    for (int m = 1; m <= 8; m <<= 1) {
      float od = __shfl_xor(bd, m, 16);
      int oi = __shfl_xor(bi, m, 16);
      if (od < bd || (od == bd && oi < bi)) { bd = od; bi = oi; }  // ties -> low idx
    }
    if ((lane & 15) == 0) {
      int M = v + ((lane < 16) ? 0 : 8);
      indices[row_base + M] = bi;
    }
  }
}

// ---------------- commit loss: 0.25 * mean((cb[idx]-vq)^2) ----------------
__global__ void __launch_bounds__(256) k_vqloss(const float* __restrict__ vq,
                                                const float* __restrict__ cb,
                                                const int* __restrict__ indices,
                                                float* __restrict__ out0) {
  int row = blockIdx.x * blockDim.x + threadIdx.x;   // 131072 exact
  int idx = indices[row];
  const float* q = cb + idx * 25;
  const float* v = vq + row * 25;
  float s = 0.f;
  for (int k = 0; k < 25; ++k) { float d = q[k] - v[k]; s += d * d; }
  for (int m = 16; m >= 1; m >>= 1) s += __shfl_xor(s, m, 32);
  if ((threadIdx.x & 31) == 0) atomicAdd(out0, s * (0.25f / 3276800.0f));
}

// ---------------- latent = gather(cb, idx) x trans_w + trans_b ----------------
__global__ void __launch_bounds__(256) k_latent(const int* __restrict__ indices,
                                                const float* __restrict__ cb,
                                                const float* __restrict__ w,
                                                const float* __restrict__ bias,
                                                float* __restrict__ latent) {
  __shared__ float zr[128 * 25];
  int b = blockIdx.x, tid = threadIdx.x;
  for (int i = tid; i < 3200; i += 256) {
    int e = i / 25, k = i % 25;
    zr[i] = cb[indices[b * 128 + e] * 25 + k];
  }
  __syncthreads();
  for (int j = tid; j < 96 * 25; j += 256) {
    int o = j / 25, p = j % 25;
    float acc = bias[o];
    for (int e = 0; e < 128; ++e) acc += zr[e * 25 + p] * w[e * 96 + o];
    latent[b * 2400 + j] = acc;
  }
}

// ---------------- pose head: relu(feat@W1+b1)@W2+b2 ----------------
__global__ void __launch_bounds__(256) k_pose(const float* __restrict__ latent,
                                              const float* __restrict__ w1,
                                              const float* __restrict__ b1,
                                              const float* __restrict__ w2,
                                              const float* __restrict__ b2,
                                              float* __restrict__ kp) {
  __shared__ float hid[32];
  int b = blockIdx.x, tid = threadIdx.x;
  int u = tid >> 3, part = tid & 7;
  const float* f = latent + b * 2400;
  float s = 0.f;
  for (int e = part * 300; e < part * 300 + 300; ++e) s += f[e] * w1[e * 32 + u];
  for (int m = 1; m <= 4; m <<= 1) s += __shfl_xor(s, m, 8);
  if (part == 0) { float h = s + b1[u]; hid[u] = h > 0.f ? h : 0.f; }
  __syncthreads();
  if (tid < 36) {
    float acc = b2[tid];
    for (int uu = 0; uu < 32; ++uu) acc += hid[uu] * w2[uu * 36 + tid];
    kp[b * 36 + tid] = acc;
  }
}

// ---------------- unpool(idx2) + conv 96->32 (3x3 pad1) + relu ----------------
__global__ void __launch_bounds__(256) k_dec1(const float* __restrict__ latent,
                                              const int* __restrict__ idx2,
                                              const float* __restrict__ w,
                                              const float* __restrict__ bias,
                                              float* __restrict__ d1) {
  __shared__ float u2[96 * 100];
  int b = blockIdx.x, tid = threadIdx.x;
  for (int i = tid; i < 9600; i += 256) u2[i] = 0.f;
  __syncthreads();
  for (int i = tid; i < 96 * 25; i += 256) {
    int o = i / 25, p = i % 25;
    int py = p / 5, px = p % 5;
    int k = idx2[b * 2400 + i];
    u2[o * 100 + (2 * py + (k >> 1)) * 10 + 2 * px + (k & 1)] = latent[b * 2400 + i];
  }
  __syncthreads();
  for (int j = tid; j < 32 * 100; j += 256) {
    int oc = j / 100, pos = j % 100;
    int y = pos / 10, x = pos % 10;
    float acc = bias[oc];
    for (int ic = 0; ic < 96; ++ic) {
      const float* sr = u2 + ic * 100;
      const float* wr = w + (oc * 96 + ic) * 9;
      for (int ky = 0; ky < 3; ++ky) {
        int yy = y + ky - 1;
        if (yy < 0 || yy >= 10) continue;
        for (int kx = 0; kx < 3; ++kx) {
          int xx = x + kx - 1;
          if (xx < 0 || xx >= 10) continue;
          acc += sr[yy * 10 + xx] * wr[ky * 3 + kx];
        }
      }
    }
    d1[b * 3200 + j] = acc > 0.f ? acc : 0.f;
  }
}

// ---------------- unpool(idx1) + conv 32->3 (3x3 pad1) ----------------
__global__ void __launch_bounds__(256) k_dec2(const float* __restrict__ d1,
                                              const int* __restrict__ idx1,
                                              const float* __restrict__ w,
                                              const float* __restrict__ bias,
                                              float* __restrict__ rx) {
  __shared__ float u1[32 * 400];
  int b = blockIdx.x, tid = threadIdx.x;
  for (int i = tid; i < 12800; i += 256) u1[i] = 0.f;
  __syncthreads();
  for (int i = tid; i < 32 * 100; i += 256) {
    int c = i / 100, p = i % 100;
    int py = p / 10, px = p % 10;
    int k = idx1[b * 3200 + i];
    u1[c * 400 + (2 * py + (k >> 1)) * 20 + 2 * px + (k & 1)] = d1[b * 3200 + i];
  }
  __syncthreads();
  for (int j = tid; j < 3 * 400; j += 256) {
    int oc = j / 400, pos = j % 400;
    int y = pos / 20, x = pos % 20;
    float acc = bias[oc];
    for (int ic = 0; ic < 32; ++ic) {
      const float* sr = u1 + ic * 400;
      const float* wr = w + (oc * 32 + ic) * 9;
      for (int ky = 0; ky < 3; ++ky) {
        int yy = y + ky - 1;
        if (yy < 0 || yy >= 20) continue;
        for (int kx = 0; kx < 3; ++kx) {
          int xx = x + kx - 1;
          if (xx < 0 || xx >= 20) continue;
          acc += sr[yy * 20 + xx] * wr[ky * 3 + kx];
        }
      }
    }
    rx[(b * 3 + oc) * 400 + pos] = acc;
  }
}

__global__ void k_zero(float* p) {
  if (threadIdx.x == 0 && blockIdx.x == 0) p[0] = 0.f;
}

extern "C" void kernel_launch(void* const* d_in, const int* in_sizes, int n_in,
                              void* d_out, int out_size, void* d_ws, size_t ws_size,
                              hipStream_t stream) {
  (void)in_sizes; (void)n_in; (void)out_size; (void)ws_size;
  const float* x   = (const float*)d_in[0];
  const float* e1w = (const float*)d_in[1];  const float* e1b = (const float*)d_in[2];
  const float* e2w = (const float*)d_in[3];  const float* e2b = (const float*)d_in[4];
  const float* pw  = (const float*)d_in[5];  const float* pb  = (const float*)d_in[6];
  const float* cb  = (const float*)d_in[7];
  const float* tw  = (const float*)d_in[8];  const float* tb  = (const float*)d_in[9];
  const float* d1w = (const float*)d_in[10]; const float* d1b = (const float*)d_in[11];
  const float* d2w = (const float*)d_in[12]; const float* d2b = (const float*)d_in[13];
  const float* hw1 = (const float*)d_in[14]; const float* hb1 = (const float*)d_in[15];
  const float* hw2 = (const float*)d_in[16]; const float* hb2 = (const float*)d_in[17];

  float* ws = (float*)d_ws;
  float* p1      = ws;                        // 3,276,800 f
  int*   idx1    = (int*)(ws + 3276800);      // 3,276,800 i
  float* z       = ws + 6553600;              // 2,457,600 f
  int*   idx2    = (int*)(ws + 9011200);      // 2,457,600 i
  float* vq      = ws + 11468800;             // 3,276,800 f
  int*   indices = (int*)(ws + 14745600);     //   131,072 i
  float* cbsq    = ws + 14876672;             //     1,024 f
  float* latent  = ws + 14877696;             // 2,457,600 f
  float* d1      = ws + 17335296;             // 3,276,800 f

  float* out = (float*)d_out;
  float* rx  = out + 1;
  float* kp  = out + 1 + 1228800;

  k_zero<<<1, 32, 0, stream>>>(out);
  k_enc1<<<12800, 256, 0, stream>>>(x, e1w, e1b, p1, idx1);
  k_enc2<<<1024, 256, 0, stream>>>(p1, e2w, e2b, z, idx2);
  k_pre<<<12800, 256, 0, stream>>>(z, pw, pb, vq);
  k_cbsq<<<4, 256, 0, stream>>>(cb, cbsq);
  k_vq_argmin<<<1024, 256, 0, stream>>>(vq, cb, cbsq, indices);
  k_vqloss<<<512, 256, 0, stream>>>(vq, cb, indices, out);
  k_latent<<<1024, 256, 0, stream>>>(indices, cb, tw, tb, latent);
  k_pose<<<1024, 256, 0, stream>>>(latent, hw1, hb1, hw2, hb2, kp);
  k_dec1<<<1024, 256, 0, stream>>>(latent, idx2, d1w, d1b, d1);
  k_dec2<<<1024, 256, 0, stream>>>(d1, idx1, d2w, d2b, rx);
}